// Linear_block_88888643158261
// MI455X (gfx1250) — compile-verified
//
#include <hip/hip_runtime.h>

typedef __attribute__((ext_vector_type(2))) float v2f;
typedef __attribute__((ext_vector_type(8))) float v8f;

#define BATCH_N     524288
#define KDIM        128   // IN_SIZE
#define NDIM        128   // OUT_SIZE
#define ROWS_PER_WG 128   // 8 waves x 16 rows
#define THREADS     256

// out[n,o] = sum_i x[n,i] * W[o,i] + b[o]
// WMMA D = A(16x4, M=x-rows, K=i) * B(4x16, K=i, N=o) + C
__global__ __launch_bounds__(THREADS)
void linear_wmma_f32(const float* __restrict__ x,
                     const float* __restrict__ W,
                     const float* __restrict__ b,
                     float* __restrict__ out)
{
    // W staged in B-fragment order:
    // lW[((nt*32+kt)*32 + L)*2 + e] = W[(nt*16 + (L&15))*128 + kt*4 + (L>>4)*2 + e]
    // -> inner-loop ds_load_b64 is contiguous per wave (lane L reads bytes [8L, 8L+8))
    __shared__ float lW[8 * 32 * 64];   // 65536 B

    const int t    = threadIdx.x;
    const int lane = t & 31;
    const int wave = t >> 5;

    // ---- stage W (64KB) into LDS, swizzled into fragment layout ----
    #pragma unroll
    for (int i = 0; i < 32; ++i) {
        const int j  = i * THREADS + t;       // b64-pair index, 0..8191
        const int nt = j >> 10;               // 0..7   (N tile)
        const int kt = (j >> 5) & 31;         // 0..31  (K step)
        const int L  = j & 31;                // lane slot
        const int col = kt * 4 + ((L >> 4) << 1);
        const v2f w2 = *(const v2f*)(W + (nt * 16 + (L & 15)) * KDIM + col);
        *(v2f*)(&lW[j * 2]) = w2;
    }
    __syncthreads();

    // ---- A fragment addressing: lane L holds row (L&15), K-half (L>>4)*2 ----
    const int row0 = blockIdx.x * ROWS_PER_WG + wave * 16;
    const int am   = lane & 15;
    const int akh  = (lane >> 4) << 1;
    const float* xrow = x + (size_t)(row0 + am) * KDIM + akh;

    v8f acc[8] = {};   // 8 N-tiles of 16x16 f32 accumulators

    #pragma unroll 4
    for (int kt = 0; kt < 32; ++kt) {
        const v2f a = *(const v2f*)(xrow + kt * 4);   // global_load_b64
        #pragma unroll
        for (int nt = 0; nt < 8; ++nt) {
            const v2f bf = *(const v2f*)(&lW[((nt * 32 + kt) * 32 + lane) * 2]); // ds_load_b64
            // (neg_a, A, neg_b, B, c_mod, C, reuse_a, reuse_b)
            acc[nt] = __builtin_amdgcn_wmma_f32_16x16x4_f32(
                false, a, false, bf, (short)0, acc[nt], false, false);
        }
    }

    // ---- bias + streaming store ----
    // D layout: VGPR r -> row (r + 8*(lane>>4)), col = lane&15 within tile
    const int cn    = lane & 15;
    const int mbase = (lane >> 4) << 3;
    #pragma unroll
    for (int nt = 0; nt < 8; ++nt) {
        const float bias = b[nt * 16 + cn];
        const int   colg = nt * 16 + cn;
        #pragma unroll
        for (int r = 0; r < 8; ++r) {
            float* dst = out + (size_t)(row0 + mbase + r) * NDIM + colg;
            __builtin_nontemporal_store(acc[nt][r] + bias, dst);
        }
    }
}

extern "C" void kernel_launch(void* const* d_in, const int* in_sizes, int n_in,
                              void* d_out, int out_size, void* d_ws, size_t ws_size,
                              hipStream_t stream) {
    const float* x = (const float*)d_in[0];
    const float* W = (const float*)d_in[1];
    const float* b = (const float*)d_in[2];
    float* out = (float*)d_out;

    dim3 grid(BATCH_N / ROWS_PER_WG);   // 4096 workgroups
    dim3 block(THREADS);                // 8 wave32 waves
    linear_wmma_f32<<<grid, block, 0, stream>>>(x, W, b, out);
}